// WaveNet_23055384445843
// MI455X (gfx1250) — compile-verified
//
#include <hip/hip_runtime.h>
#include <hip/hip_bf16.h>

// ---------------------------------------------------------------------------
// Fused WaveNet inference for MI455X (gfx1250, wave32).
//
// - Only t = L-1 of the output matters => compute only the t in [4096, 8192)
//   suffix (receptive field 4095), with a per-layer validity frontier tmin.
// - h ping-pong buffers: 2 x 16 MB, fully L2-resident (192 MB L2); workload
//   is L2-request-rate bound => h stored channel-PAIR interleaved
//   [b][ch/2][t][2] so every h access is one b64 op on a contiguous line.
// - All GEMMs on the matrix pipe: V_WMMA_F32_16X16X4_F32 (exact fp32,
//   K=32 as 8 chained K=4 steps).
// - Exact per-layer grids: ceil((L - tmin)/64) time-blocks at the tail.
// ---------------------------------------------------------------------------

typedef __attribute__((ext_vector_type(2))) float v2f;
typedef __attribute__((ext_vector_type(8))) float v8f;

#define BATCH 32
#define LFULL 8192
#define FIN   16
#define RCH   32
#define TL    4096          // stored time suffix length
#define T0    (LFULL - TL)  // 4096
#define WST   34            // padded LDS stride (floats) for weight rows
#define ZPR   66            // padded pair-stride for zbuf rows (pairs)

// float index of (channel-pair c2, time t) in the interleaved h layout
#define HIDX(c2, t) ((((c2) * TL) + (t)) * 2)

// D = A(16x4) * B(4x16) + C, fp32.
// A-frag: lanes 0-15 hold M=lane, {v0,v1}={K0,K1}; lanes 16-31: {K2,K3}.
// C/D: VGPR r, lane n -> M = r + 8*(n>=16), N = n%16.
__device__ __forceinline__ v8f wmma4(v2f a, v2f b, v8f c) {
    return __builtin_amdgcn_wmma_f32_16x16x4_f32(false, a, false, b,
                                                 (short)0, c, false, false);
}

// ---------------------------------------------------------------------------
// Kernel A: input 1x1 conv  x(B,L,F) -> h suffix (pair-interleaved), and
// zero the skip accumulator. Grid (TL/64, B), 128 threads = 4 waves.
// ---------------------------------------------------------------------------
__global__ void wn_in_conv(const float* __restrict__ x,
                           const float* __restrict__ w,   // (R,F)
                           const float* __restrict__ bias,
                           float* __restrict__ h,
                           float* __restrict__ skip_acc) {
    int tid = threadIdx.x;
    if (blockIdx.x == 0 && blockIdx.y == 0) {
        for (int i = tid; i < BATCH * RCH; i += 128) skip_acc[i] = 0.f;
    }
    int b    = blockIdx.y;
    int lane = tid & 31, wave = tid >> 5;
    int hi   = lane >> 4;
    int tloc = wave * 16 + (lane & 15);
    int t    = T0 + blockIdx.x * 64 + tloc;

    const float* xb = x + (size_t)b * LFULL * FIN;

    // B-frags: K = input feature, N = time (x is feature-major => b64 loads)
    v2f bf[4];
#pragma unroll
    for (int k = 0; k < 4; ++k) {
        int ci = 4 * k + 2 * hi;
        bf[k] = *(const v2f*)&xb[t * FIN + ci];
    }

    float* hb = h + (size_t)b * RCH * TL;
    int tl = t - T0;
#pragma unroll
    for (int mt = 0; mt < 2; ++mt) {
        v8f acc;
#pragma unroll
        for (int r = 0; r < 8; ++r) acc[r] = bias[mt * 16 + hi * 8 + r];
#pragma unroll
        for (int k = 0; k < 4; ++k) {
            int ci = 4 * k + 2 * hi;
            v2f a = *(const v2f*)&w[(mt * 16 + (lane & 15)) * FIN + ci];
            acc = wmma4(a, bf[k], acc);
        }
        // pair-interleaved store: (r, r+1) are adjacent channels
#pragma unroll
        for (int r2 = 0; r2 < 4; ++r2) {
            int c2 = mt * 8 + hi * 4 + r2;
            v2f o; o.x = acc[2 * r2]; o.y = acc[2 * r2 + 1];
            *(v2f*)&hb[HIDX(c2, tl)] = o;
        }
    }
}

// ---------------------------------------------------------------------------
// Kernel B: one dilated gated layer, fused:
//   y = W0*h[t-d/2] + W1*h[t+d/2] + b      (64 ch, 4 WMMA M-tiles)
//   z = tanh(y[32:]) * sigmoid(y[:32])
//   skip_acc[b] += Wskip*z[t=L-1] + bskip  (only block owning t=L-1)
//   h_out = Wres*z + bres + h_in           (2 WMMA M-tiles)
// Grid.x = exact number of 64-wide tail tiles; stores masked below tmin.
// ---------------------------------------------------------------------------
__global__ void wn_layer(const float* __restrict__ h_in,
                         float* __restrict__ h_out,
                         const float* __restrict__ dw,  // (64,32,2)
                         const float* __restrict__ db,  // (64)
                         const float* __restrict__ sw,  // (32,32)
                         const float* __restrict__ sb,  // (32)
                         const float* __restrict__ rw,  // (32,32)
                         const float* __restrict__ rb,  // (32)
                         float* __restrict__ skip_acc,
                         int halfd, int tmin) {
    __shared__ __align__(16) float sdw[2 * 64 * WST];    // [tap][co][ci]
    __shared__ __align__(16) float srw[RCH * WST];
    __shared__ __align__(16) float ssw[RCH * RCH];
    __shared__ __align__(16) float zbuf[16 * ZPR * 2];   // [c2][tloc][2], padded

    int tid = threadIdx.x;
    int b   = blockIdx.y;
    // tail-positioned tile: last block always owns t = L-1
    int tileIdx  = (TL / 64) - gridDim.x + blockIdx.x;
    int tileBase = T0 + tileIdx * 64;

    // Stage weights into LDS (padded strides; float2-aligned rows).
    for (int i = tid; i < 64 * RCH * 2; i += 128) {
        int tap = i & 1, tmp = i >> 1, ci = tmp & 31, co = tmp >> 5;
        sdw[(tap * 64 + co) * WST + ci] = dw[i];
    }
    for (int i = tid; i < RCH * RCH; i += 128) {
        srw[(i >> 5) * WST + (i & 31)] = rw[i];
        ssw[i] = sw[i];
    }
    __syncthreads();

    int lane = tid & 31, wave = tid >> 5;
    int hi   = lane >> 4;
    int tloc = wave * 16 + (lane & 15);
    int t    = tileBase + tloc;
    int tl   = t - T0;

    const float* hb = h_in + (size_t)b * RCH * TL;

    // B-frags for both taps. For valid columns t-d/2 is always in-buffer;
    // t+d/2 past the end is the conv's genuine zero padding.
    int tlm = tl - halfd; if (tlm < 0) tlm = 0;        // garbage cols only; masked later
    int tlp = tl + halfd; bool okp = (tlp < TL); if (!okp) tlp = 0;
    v2f zero; zero.x = 0.f; zero.y = 0.f;
    v2f bm[8], bp[8];
#pragma unroll
    for (int k = 0; k < 8; ++k) {
        int c2 = 2 * k + hi;                            // ci = 4k + 2*hi
        bm[k] = *(const v2f*)&hb[HIDX(c2, tlm)];
        bp[k] = okp ? *(const v2f*)&hb[HIDX(c2, tlp)] : zero;
    }

    // Residual values (C/D layout, channel pairs => b64 loads).
    v2f hres[2][4];
#pragma unroll
    for (int mt = 0; mt < 2; ++mt)
#pragma unroll
        for (int r2 = 0; r2 < 4; ++r2)
            hres[mt][r2] = *(const v2f*)&hb[HIDX(mt * 8 + hi * 4 + r2, tl)];

    // Dilated conv: 4 output M-tiles (gate = ch 0..31, filter = ch 32..63).
    v8f y[4];
#pragma unroll
    for (int mt = 0; mt < 4; ++mt) {
        v8f acc;
#pragma unroll
        for (int r = 0; r < 8; ++r) acc[r] = db[mt * 16 + hi * 8 + r];
#pragma unroll
        for (int k = 0; k < 8; ++k) {
            int ci  = 4 * k + 2 * hi;
            int row = (mt * 16 + (lane & 15)) * WST + ci;
            v2f a0 = *(const v2f*)&sdw[row];              // tap 0
            acc = wmma4(a0, bm[k], acc);
            v2f a1 = *(const v2f*)&sdw[64 * WST + row];   // tap 1
            acc = wmma4(a1, bp[k], acc);
        }
        y[mt] = acc;
    }

    // Gated activation; reshape D-layout -> pair-interleaved [c2][t] in LDS.
#pragma unroll
    for (int mt = 0; mt < 2; ++mt) {
#pragma unroll
        for (int r2 = 0; r2 < 4; ++r2) {
            float g0 = y[mt][2 * r2],     f0 = y[mt + 2][2 * r2];
            float g1 = y[mt][2 * r2 + 1], f1 = y[mt + 2][2 * r2 + 1];
            v2f zz;
            zz.x = tanhf(f0) * (1.f / (1.f + __expf(-g0)));
            zz.y = tanhf(f1) * (1.f / (1.f + __expf(-g1)));
            int c2 = mt * 8 + hi * 4 + r2;
            *(v2f*)&zbuf[(c2 * ZPR + tloc) * 2] = zz;
        }
    }
    __syncthreads();

    // Skip path: only the final timestep ever reaches the output head.
    if (blockIdx.x == gridDim.x - 1 && tid < RCH) {
        float acc = sb[tid];
#pragma unroll
        for (int ci = 0; ci < RCH; ++ci)
            acc += ssw[tid * RCH + ci] * zbuf[((ci >> 1) * ZPR + 63) * 2 + (ci & 1)];
        skip_acc[b * RCH + tid] += acc;  // single writer per (b, layer)
    }

    // Residual 1x1 conv + residual add.
    v2f bz[8];
#pragma unroll
    for (int k = 0; k < 8; ++k) {
        int c2 = 2 * k + hi;
        bz[k] = *(const v2f*)&zbuf[(c2 * ZPR + tloc) * 2];
    }
    float* ho = h_out + (size_t)b * RCH * TL;
    bool wr = (t >= tmin);
#pragma unroll
    for (int mt = 0; mt < 2; ++mt) {
        v8f acc;
#pragma unroll
        for (int r = 0; r < 8; ++r) acc[r] = rb[mt * 16 + hi * 8 + r];
#pragma unroll
        for (int k = 0; k < 8; ++k) {
            int ci = 4 * k + 2 * hi;
            v2f a = *(const v2f*)&srw[(mt * 16 + (lane & 15)) * WST + ci];
            acc = wmma4(a, bz[k], acc);
        }
        if (wr) {
#pragma unroll
            for (int r2 = 0; r2 < 4; ++r2) {
                v2f o;
                o.x = acc[2 * r2]     + hres[mt][r2].x;
                o.y = acc[2 * r2 + 1] + hres[mt][r2].y;
                *(v2f*)&ho[HIDX(mt * 8 + hi * 4 + r2, tl)] = o;
            }
        }
    }
}

// ---------------------------------------------------------------------------
// Kernel C: output head at t = L-1 only. One wave32, lane = batch index.
// ---------------------------------------------------------------------------
__global__ void wn_head(const float* __restrict__ skip_acc,
                        const float* __restrict__ w1, const float* __restrict__ b1,
                        const float* __restrict__ w2, const float* __restrict__ b2,
                        float* __restrict__ out) {
    int b = threadIdx.x;
    if (b >= BATCH) return;
    float s[RCH];
#pragma unroll
    for (int c = 0; c < RCH; ++c) {
        float v = skip_acc[b * RCH + c];
        s[c] = v > 0.f ? v : 0.f;
    }
    float o = b2[0];
    for (int j = 0; j < RCH; ++j) {
        float a = b1[j];
#pragma unroll
        for (int c = 0; c < RCH; ++c) a += w1[j * RCH + c] * s[c];
        o += w2[j] * (a > 0.f ? a : 0.f);
    }
    out[b] = o;
}

// ---------------------------------------------------------------------------
extern "C" void kernel_launch(void* const* d_in, const int* in_sizes, int n_in,
                              void* d_out, int out_size, void* d_ws, size_t ws_size,
                              hipStream_t stream) {
    const float* x     = (const float*)d_in[0];
    const float* in_w  = (const float*)d_in[1];
    const float* in_b  = (const float*)d_in[2];
    const float* dil_w = (const float*)d_in[3];
    const float* dil_b = (const float*)d_in[4];
    const float* sk_w  = (const float*)d_in[5];
    const float* sk_b  = (const float*)d_in[6];
    const float* rs_w  = (const float*)d_in[7];
    const float* rs_b  = (const float*)d_in[8];
    const float* f1_w  = (const float*)d_in[9];
    const float* f1_b  = (const float*)d_in[10];
    const float* f2_w  = (const float*)d_in[11];
    const float* f2_b  = (const float*)d_in[12];

    // Workspace: two 16 MB suffix ping-pong h buffers + 4 KB skip accumulator.
    float* hA       = (float*)d_ws;
    float* hB       = hA + (size_t)BATCH * RCH * TL;
    float* skip_acc = hB + (size_t)BATCH * RCH * TL;

    dim3 blk(128);
    wn_in_conv<<<dim3(TL / 64, BATCH), blk, 0, stream>>>(x, in_w, in_b, hA, skip_acc);

    static const int dil[12] = {2, 4, 8, 16, 32, 64, 128, 256, 512, 1024, 2048, 4096};
    float* hin = hA;
    float* hout = hB;
    int tmin = T0;
    for (int l = 0; l < 12; ++l) {
        int hd = dil[l] >> 1;
        tmin += hd;                            // validity frontier advances by d/2
        int nblkx = (LFULL - tmin + 63) / 64;  // exact tail coverage
        wn_layer<<<dim3(nblkx, BATCH), blk, 0, stream>>>(hin, hout,
                                           dil_w + l * 64 * 32 * 2, dil_b + l * 64,
                                           sk_w + l * 32 * 32, sk_b + l * 32,
                                           rs_w + l * 32 * 32, rs_b + l * 32,
                                           skip_acc, hd, tmin);
        float* tp = hin; hin = hout; hout = tp;
    }
    wn_head<<<1, 32, 0, stream>>>(skip_acc, f1_w, f1_b, f2_w, f2_b, (float*)d_out);
}